// GATEModel_21440476742178
// MI455X (gfx1250) — compile-verified
//
#include <hip/hip_runtime.h>
#include <hip/hip_bf16.h>

// ---------------------------------------------------------------------------
// GATE graph auto-encoder for MI455X (gfx1250, wave32).
// Dense GEMMs use V_WMMA_F32_16X16X4_F32 (f32-exact matrix path).
// Edge attention/aggregation uses L2-resident gathers + f32 atomics.
// ---------------------------------------------------------------------------

typedef float v2f __attribute__((ext_vector_type(2)));
typedef float v8f __attribute__((ext_vector_type(8)));

#define N_NODES 50000
#define N_EDGES 800000
#define D_IN    128
#define D_HID   128
#define D_OUT   64

// ---------------- fill ----------------
__global__ void k_fill(float* __restrict__ p, long long n, float v) {
    long long i = (long long)blockIdx.x * blockDim.x + threadIdx.x;
    if (i < n) p[i] = v;
}

// ---------------- GEMM: P = X @ W^T  (W is [nout, K] row-major) ----------------
// One wave computes a 16(M) x 16(N) tile; K stepped by 4 via WMMA f32 16x16x4.
// grid.x = nrows/16 (= 3125), blockDim.x = 32 * (nout/16).
__global__ void k_gemm_xWT(const float* __restrict__ X, const float* __restrict__ W,
                           float* __restrict__ P, int K, int nout) {
    const int lane = threadIdx.x & 31;
    const int wave = threadIdx.x >> 5;
    const int mBase = blockIdx.x << 4;
    const int nBase = wave << 4;
    if (nBase >= nout) return;
    const int half = lane >> 4;       // 0: K pair {0,1}; 1: K pair {2,3}
    const int l16  = lane & 15;
    const float* __restrict__ xr = X + (long long)(mBase + l16) * K;  // A row
    const float* __restrict__ wr = W + (long long)(nBase + l16) * K;  // B col (row of W)
    v8f acc = {};
    for (int k = 0; k < K; k += 4) {
        const int ka = k + 2 * half;
        v2f a; a.x = xr[ka]; a.y = xr[ka + 1];
        v2f b; b.x = wr[ka]; b.y = wr[ka + 1];
        acc = __builtin_amdgcn_wmma_f32_16x16x4_f32(false, a, false, b,
                                                    (short)0, acc, false, false);
    }
    const int col = nBase + l16;
    const int rowOff = mBase + (half << 3);
#pragma unroll
    for (int j = 0; j < 8; ++j)
        P[(long long)(rowOff + j) * nout + col] = acc[j];
}

// ---------------- GEMM: P = X @ W  (W is [K, nout] row-major) ----------------
__global__ void k_gemm_xW(const float* __restrict__ X, const float* __restrict__ W,
                          float* __restrict__ P, int K, int nout) {
    const int lane = threadIdx.x & 31;
    const int wave = threadIdx.x >> 5;
    const int mBase = blockIdx.x << 4;
    const int nBase = wave << 4;
    if (nBase >= nout) return;
    const int half = lane >> 4;
    const int l16  = lane & 15;
    const float* __restrict__ xr = X + (long long)(mBase + l16) * K;
    const int bCol = nBase + l16;
    v8f acc = {};
    for (int k = 0; k < K; k += 4) {
        const int ka = k + 2 * half;
        v2f a; a.x = xr[ka]; a.y = xr[ka + 1];
        v2f b; b.x = W[(long long)ka * nout + bCol];
               b.y = W[(long long)(ka + 1) * nout + bCol];
        acc = __builtin_amdgcn_wmma_f32_16x16x4_f32(false, a, false, b,
                                                    (short)0, acc, false, false);
    }
    const int rowOff = mBase + (half << 3);
#pragma unroll
    for (int j = 0; j < 8; ++j)
        P[(long long)(rowOff + j) * nout + bCol] = acc[j];
}

// ---------------- a0[n] = P[n,:]·v0 ; a1[n] = P[n,:]·v1 (wave per row) ----------------
__global__ void k_row_dots(const float* __restrict__ P, const float* __restrict__ v0,
                           const float* __restrict__ v1, float* __restrict__ a0,
                           float* __restrict__ a1, int n, int D) {
    const int w = blockIdx.x * (blockDim.x >> 5) + (threadIdx.x >> 5);
    const int lane = threadIdx.x & 31;
    if (w >= n) return;
    const float* __restrict__ p = P + (long long)w * D;
    float s0 = 0.f, s1 = 0.f;
    for (int i = lane; i < D; i += 32) {
        const float pv = p[i];
        s0 += pv * v0[i];
        s1 += pv * v1[i];
    }
#pragma unroll
    for (int off = 16; off; off >>= 1) {
        s0 += __shfl_down(s0, off, 32);
        s1 += __shfl_down(s1, off, 32);
    }
    if (lane == 0) { a0[w] = s0; a1[w] = s1; }
}

// ---------------- scores = sigmoid(a0[row]+a1[col]); segmax by row ----------------
// Scores are in (0,1): positive floats order like their uint bit patterns.
__global__ void k_edge_scores_max(const float* __restrict__ a0, const float* __restrict__ a1,
                                  const int* __restrict__ row, const int* __restrict__ col,
                                  float* __restrict__ scores, float* __restrict__ segmax,
                                  int nE) {
    const int e = blockIdx.x * blockDim.x + threadIdx.x;
    if (e >= nE) return;
    const int r = row[e];
    const float z = a0[r] + a1[col[e]];
    const float s = 1.f / (1.f + __expf(-z));
    scores[e] = s;
    atomicMax((unsigned int*)&segmax[r], __float_as_uint(s));
}

// ---------------- e = exp(score - max[row]); segsum by row ----------------
__global__ void k_edge_exp_sum(float* __restrict__ attn, const float* __restrict__ segmax,
                               const int* __restrict__ row, float* __restrict__ dsum, int nE) {
    const int e = blockIdx.x * blockDim.x + threadIdx.x;
    if (e >= nE) return;
    const int r = row[e];
    const float ev = __expf(attn[e] - segmax[r]);
    attn[e] = ev;
    atomicAdd(&dsum[r], ev);
}

// ---------------- attn = e / (sum[row] + 1e-16) ----------------
__global__ void k_edge_norm(float* __restrict__ attn, const float* __restrict__ dsum,
                            const int* __restrict__ row, int nE) {
    const int e = blockIdx.x * blockDim.x + threadIdx.x;
    if (e >= nE) return;
    attn[e] = attn[e] / (dsum[row[e]] + 1e-16f);
}

// ---------------- out[row] += src[col] * attn[e]  (wave per edge) ----------------
__global__ void k_edge_aggregate(const float* __restrict__ src, const int* __restrict__ row,
                                 const int* __restrict__ col, const float* __restrict__ attn,
                                 float* __restrict__ out, int nE, int D) {
    const int w = blockIdx.x * (blockDim.x >> 5) + (threadIdx.x >> 5);
    const int lane = threadIdx.x & 31;
    if (w >= nE) return;
    const int r = row[w], c = col[w];
    const float a = attn[w];
    const float* __restrict__ s = src + (long long)c * D;
    float* __restrict__ o = out + (long long)r * D;
    for (int i = lane; i < D; i += 32)
        atomicAdd(&o[i], s[i] * a);
}

// ---------------- sum((x - xr)^2) ----------------
__global__ void k_feature_loss(const float* __restrict__ x, const float* __restrict__ xr,
                               float* __restrict__ accum, long long n) {
    const long long i = (long long)blockIdx.x * blockDim.x + threadIdx.x;
    float v = 0.f;
    if (i < n) { const float d = x[i] - xr[i]; v = d * d; }
#pragma unroll
    for (int off = 16; off; off >>= 1) v += __shfl_down(v, off, 32);
    if ((threadIdx.x & 31) == 0) atomicAdd(accum, v);
}

// ---------------- -sum log(sigmoid(h[srow]·h[scol]) + 1e-8)  (wave per pair) ----------------
__global__ void k_structure_loss(const float* __restrict__ h, const int* __restrict__ srow,
                                 const int* __restrict__ scol, float* __restrict__ accum,
                                 int nP, int D) {
    const int w = blockIdx.x * (blockDim.x >> 5) + (threadIdx.x >> 5);
    const int lane = threadIdx.x & 31;
    if (w >= nP) return;
    const float* __restrict__ ha = h + (long long)srow[w] * D;
    const float* __restrict__ hb = h + (long long)scol[w] * D;
    float dot = 0.f;
    for (int i = lane; i < D; i += 32) dot += ha[i] * hb[i];
#pragma unroll
    for (int off = 16; off; off >>= 1) dot += __shfl_down(dot, off, 32);
    if (lane == 0) {
        const float sg = 1.f / (1.f + __expf(-dot));
        atomicAdd(accum, -logf(sg + 1e-8f));
    }
}

// ---------------- loss = sqrt(fs) + ss ----------------
__global__ void k_finalize(const float* __restrict__ accum, float* __restrict__ out0) {
    out0[0] = sqrtf(accum[0]) + accum[1];
}

// ---------------------------------------------------------------------------
extern "C" void kernel_launch(void* const* d_in, const int* in_sizes, int n_in,
                              void* d_out, int out_size, void* d_ws, size_t ws_size,
                              hipStream_t stream) {
    (void)in_sizes; (void)n_in; (void)out_size; (void)ws_size;
    const float* x   = (const float*)d_in[0];
    const int*   ei  = (const int*)d_in[1];
    const int*   sp  = (const int*)d_in[2];
    const float* W0  = (const float*)d_in[3];   // [128,128]
    const float* v00 = (const float*)d_in[4];
    const float* v10 = (const float*)d_in[5];
    const float* W1  = (const float*)d_in[6];   // [64,128]
    const float* v01 = (const float*)d_in[7];
    const float* v11 = (const float*)d_in[8];

    const int* row = ei;            // edge_index[0]
    const int* col = ei + N_EDGES;  // edge_index[1]
    const int* srow = sp;
    const int* scol = sp + N_EDGES;

    float* out   = (float*)d_out;
    float* h_out = out + 1;                                // [N, 64]
    float* xrec  = out + 1 + (long long)N_NODES * D_OUT;   // [N, 128]

    // ----- workspace carve-out (~71 MB) -----
    float* w = (float*)d_ws;
    float* bufA  = w; w += (long long)N_NODES * 128;  // pre-activations / reused
    float* bufB  = w; w += (long long)N_NODES * 128;  // h1 / d
    float* bufC  = w; w += (long long)N_NODES * 64;   // h2_pre
    float* a0    = w; w += N_NODES;
    float* a1    = w; w += N_NODES;
    float* segm  = w; w += N_NODES;                   // segment max (+ dsum adjacent)
    float* dsum  = w; w += N_NODES;
    float* attn0 = w; w += N_EDGES;
    float* attn1 = w; w += N_EDGES;
    float* accum = w; w += 2;

    const int mTiles   = N_NODES / 16;                 // 3125 (exact)
    const int eBlocks1 = (N_EDGES + 255) / 256;        // thread-per-edge kernels
    const int eBlocksW = (N_EDGES + 7) / 8;            // wave-per-edge kernels
    const int nBlocksW = (N_NODES + 7) / 8;            // wave-per-row kernels
    const long long nFeat = (long long)N_NODES * 128;

    // ===== encoder layer 0: h1_pre = x @ W0^T =====
    k_gemm_xWT<<<mTiles, 32 * (D_HID / 16), 0, stream>>>(x, W0, bufA, D_IN, D_HID);
    k_row_dots<<<nBlocksW, 256, 0, stream>>>(bufA, v00, v10, a0, a1, N_NODES, D_HID);
    k_fill<<<(2 * N_NODES + 255) / 256, 256, 0, stream>>>(segm, 2 * N_NODES, 0.f); // segm+dsum
    k_edge_scores_max<<<eBlocks1, 256, 0, stream>>>(a0, a1, row, col, attn0, segm, N_EDGES);
    k_edge_exp_sum<<<eBlocks1, 256, 0, stream>>>(attn0, segm, row, dsum, N_EDGES);
    k_edge_norm<<<eBlocks1, 256, 0, stream>>>(attn0, dsum, row, N_EDGES);
    k_fill<<<(int)((nFeat + 255) / 256), 256, 0, stream>>>(bufB, nFeat, 0.f);
    k_edge_aggregate<<<eBlocksW, 256, 0, stream>>>(bufA, row, col, attn0, bufB, N_EDGES, D_HID);

    // ===== encoder layer 1: h2_pre = h1 @ W1^T =====
    k_gemm_xWT<<<mTiles, 32 * (D_OUT / 16), 0, stream>>>(bufB, W1, bufC, D_HID, D_OUT);
    k_row_dots<<<nBlocksW, 256, 0, stream>>>(bufC, v01, v11, a0, a1, N_NODES, D_OUT);
    k_fill<<<(2 * N_NODES + 255) / 256, 256, 0, stream>>>(segm, 2 * N_NODES, 0.f);
    k_edge_scores_max<<<eBlocks1, 256, 0, stream>>>(a0, a1, row, col, attn1, segm, N_EDGES);
    k_edge_exp_sum<<<eBlocks1, 256, 0, stream>>>(attn1, segm, row, dsum, N_EDGES);
    k_edge_norm<<<eBlocks1, 256, 0, stream>>>(attn1, dsum, row, N_EDGES);
    k_fill<<<(int)(((long long)N_NODES * D_OUT + 255) / 256), 256, 0, stream>>>(
        h_out, (long long)N_NODES * D_OUT, 0.f);
    k_edge_aggregate<<<eBlocksW, 256, 0, stream>>>(bufC, row, col, attn1, h_out, N_EDGES, D_OUT);

    // ===== decoder layer 1: d = agg(h @ W1, attn1) =====
    k_gemm_xW<<<mTiles, 32 * (D_HID / 16), 0, stream>>>(h_out, W1, bufA, D_OUT, D_HID);
    k_fill<<<(int)((nFeat + 255) / 256), 256, 0, stream>>>(bufB, nFeat, 0.f);
    k_edge_aggregate<<<eBlocksW, 256, 0, stream>>>(bufA, row, col, attn1, bufB, N_EDGES, D_HID);

    // ===== decoder layer 0: x_recon = agg(d @ W0, attn0) =====
    k_gemm_xW<<<mTiles, 32 * (D_IN / 16), 0, stream>>>(bufB, W0, bufA, D_HID, D_IN);
    k_fill<<<(int)((nFeat + 255) / 256), 256, 0, stream>>>(xrec, nFeat, 0.f);
    k_edge_aggregate<<<eBlocksW, 256, 0, stream>>>(bufA, row, col, attn0, xrec, N_EDGES, D_IN);

    // ===== losses =====
    k_fill<<<1, 32, 0, stream>>>(accum, 2, 0.f);
    k_feature_loss<<<(int)((nFeat + 255) / 256), 256, 0, stream>>>(x, xrec, accum + 0, nFeat);
    k_structure_loss<<<eBlocksW, 256, 0, stream>>>(h_out, srow, scol, accum + 1, N_EDGES, D_OUT);
    k_finalize<<<1, 1, 0, stream>>>(accum, out);
}